// GraphClassifier_72284299592019
// MI455X (gfx1250) — compile-verified
//
#include <hip/hip_runtime.h>
#include <hip/hip_bf16.h>

typedef float v2f __attribute__((ext_vector_type(2)));
typedef float v8f __attribute__((ext_vector_type(8)));

#define FEAT 128          // IN_DIM == HID == 128
#define NCLS 10

// ---------------------------------------------------------------- degree ----
__global__ __launch_bounds__(256) void k_init_deg(float* deg, int n) {
    int i = blockIdx.x * 256 + threadIdx.x;
    if (i < n) deg[i] = 1.0f;                 // self loop
}

__global__ __launch_bounds__(256) void k_deg_count(float* deg, const int* dst, int E) {
    int e = blockIdx.x * 256 + threadIdx.x;
    if (e < E) atomicAdd(&deg[dst[e]], 1.0f);
}

__global__ __launch_bounds__(256) void k_deg_to_dinv(float* deg, int n) {
    int i = blockIdx.x * 256 + threadIdx.x;
    if (i < n) { float d = deg[i]; deg[i] = (d > 0.0f) ? rsqrtf(d) : 0.0f; }
}

// ------------------------------------------------------- fp32 WMMA GEMM -----
// Y[M x 128] = X[M x 128] @ W[128 x 128], exact fp32 via V_WMMA_F32_16X16X4_F32.
// Persistent blocks: 8 waves/block, wave w owns N-tile [16w,16w+16).
// B fragments for the whole K=128 are register-resident (32 x v2f = 64 VGPRs
// per lane), loaded once per block; inner loop is ds_load_b64 + v_wmma only.
// A tile staged in LDS with pitch 132 floats (4-bank rotation -> conflict-free
// ds_load_2addr_b64 fragment fetches). Next A tile prefetched via
// global_prefetch_b8 while the current tile's WMMAs execute.
#define LDA_S 132
__global__ __launch_bounds__(256) void k_gemm_f32_wmma(const float* __restrict__ X,
                                                       const float* __restrict__ W,
                                                       float* __restrict__ Y,
                                                       int M, int Mtiles) {
    __shared__ __align__(16) float As[16 * LDA_S];

    const int wave  = threadIdx.x >> 5;       // 0..7 -> N tile
    const int lane  = threadIdx.x & 31;
    const int tileN = wave * 16;
    const int m     = lane & 15;              // A row within tile
    const int n     = lane & 15;              // B/D column within tile
    const int koff  = (lane >> 4) * 2;        // K sub-offset per ISA A/B layout
    const int rbase = (lane >> 4) * 8;        // D row base per ISA C/D layout

    // register-resident B fragments: breg[t] covers K = 4t+koff .. 4t+koff+1
    v2f breg[32];
#pragma unroll
    for (int t = 0; t < 32; ++t) {
        int k = 4 * t + koff;
        breg[t].x = W[(size_t)k       * FEAT + tileN + n];
        breg[t].y = W[(size_t)(k + 1) * FEAT + tileN + n];
    }

    for (int tile = blockIdx.x; tile < Mtiles; tile += gridDim.x) {
        const int tileM = tile * 16;

        __syncthreads();   // previous iteration finished reading As
        for (int idx = threadIdx.x; idx < 16 * FEAT; idx += 256) {
            int r = idx >> 7, cc = idx & 127;
            int grow = tileM + r;
            As[r * LDA_S + cc] = (grow < M) ? X[(size_t)grow * FEAT + cc] : 0.0f;
        }
        __syncthreads();

        // prefetch next A tile (8 KB = 256 threads x 32 B) into cache
        int ptile = tile + gridDim.x;
        if (ptile < Mtiles) {
            const char* p = (const char*)(X + (size_t)ptile * 16 * FEAT) + threadIdx.x * 32;
            __builtin_prefetch(p, 0, 0);
        }

        v8f c = {};
#pragma unroll
        for (int t = 0; t < 32; ++t) {
            v2f a = *reinterpret_cast<const v2f*>(&As[m * LDA_S + 4 * t + koff]);
            c = __builtin_amdgcn_wmma_f32_16x16x4_f32(false, a, false, breg[t],
                                                      (short)0, c, false, false);
        }

#pragma unroll
        for (int i = 0; i < 8; ++i) {
            int row = tileM + rbase + i;
            if (row < M) Y[(size_t)row * FEAT + tileN + n] = c[i];
        }
    }
}

// ---------------------------------------------------- GCN aggregation -------
// agg[i,:] = dinv[i]^2 * xw[i,:]   (self-loop term; also initializes agg)
__global__ __launch_bounds__(256) void k_self_init(float* __restrict__ agg,
                                                   const float* __restrict__ xw,
                                                   const float* __restrict__ dinv, int n) {
    long long i = (long long)blockIdx.x * 256 + threadIdx.x;
    if (i < (long long)n * FEAT) {
        int node = (int)(i >> 7);
        float w = dinv[node];
        agg[i] = w * w * xw[i];
    }
}

// one wave32 per edge: float4 gather of src row, scaled atomic scatter to dst
__global__ __launch_bounds__(256) void k_edge_agg(float* __restrict__ agg,
                                                  const float* __restrict__ xw,
                                                  const float* __restrict__ dinv,
                                                  const int* __restrict__ src,
                                                  const int* __restrict__ dst, int E) {
    long long wid = ((long long)blockIdx.x * 256 + threadIdx.x) >> 5;
    int lane = threadIdx.x & 31;
    if (wid >= E) return;
    int s = src[wid], d = dst[wid];
    float w = dinv[s] * dinv[d];
    const float4* xs = reinterpret_cast<const float4*>(xw + (size_t)s * FEAT);
    float* ad = agg + (size_t)d * FEAT;
    float4 v = xs[lane];                      // 32 lanes x 4 floats = 128
    atomicAdd(&ad[lane * 4 + 0], w * v.x);
    atomicAdd(&ad[lane * 4 + 1], w * v.y);
    atomicAdd(&ad[lane * 4 + 2], w * v.z);
    atomicAdd(&ad[lane * 4 + 3], w * v.w);
}

__global__ __launch_bounds__(256) void k_bias_act(float* h, const float* __restrict__ b,
                                                  int n, int relu) {
    long long i = (long long)blockIdx.x * 256 + threadIdx.x;
    if (i < (long long)n * FEAT) {
        float v = h[i] + b[i & 127];
        h[i] = relu ? fmaxf(v, 0.0f) : v;
    }
}

// ------------------------------------------------------------- pooling ------
__global__ __launch_bounds__(256) void k_pool_zero(float* psum, float* pcnt, int G) {
    int i = blockIdx.x * 256 + threadIdx.x;
    if (i < G * FEAT) psum[i] = 0.0f;
    if (i < G)        pcnt[i] = 0.0f;
}

__global__ __launch_bounds__(256) void k_pool_accum(float* __restrict__ psum,
                                                    float* __restrict__ pcnt,
                                                    const float* __restrict__ h,
                                                    const int* __restrict__ batch, int n) {
    long long i = (long long)blockIdx.x * 256 + threadIdx.x;
    if (i < (long long)n * FEAT) {
        int node = (int)(i >> 7), f = (int)(i & 127);
        int g = batch[node];
        atomicAdd(&psum[(size_t)g * FEAT + f], h[i]);
        if (f == 0) atomicAdd(&pcnt[g], 1.0f);
    }
}

__global__ __launch_bounds__(256) void k_classify(const float* __restrict__ psum,
                                                  const float* __restrict__ pcnt,
                                                  const float* __restrict__ Wc,
                                                  const float* __restrict__ bc,
                                                  float* __restrict__ out, int G) {
    int i = blockIdx.x * 256 + threadIdx.x;
    if (i < G * NCLS) {
        int g = i / NCLS, c = i % NCLS;
        float inv = 1.0f / fmaxf(pcnt[g], 1.0f);
        float acc = bc[c];
#pragma unroll 8
        for (int k = 0; k < FEAT; ++k)
            acc += psum[(size_t)g * FEAT + k] * inv * Wc[(size_t)k * NCLS + c];
        out[i] = acc;
    }
}

// ------------------------------------------------------------- driver -------
extern "C" void kernel_launch(void* const* d_in, const int* in_sizes, int n_in,
                              void* d_out, int out_size, void* d_ws, size_t ws_size,
                              hipStream_t stream) {
    const float* x     = (const float*)d_in[0];
    const int*   ei    = (const int*)  d_in[1];   // [2, E] flat
    const int*   batch = (const int*)  d_in[2];
    const float* W1    = (const float*)d_in[3];
    const float* b1    = (const float*)d_in[4];
    const float* W2    = (const float*)d_in[5];
    const float* b2    = (const float*)d_in[6];
    const float* Wc    = (const float*)d_in[7];
    const float* bc    = (const float*)d_in[8];
    float* out = (float*)d_out;

    const int M = in_sizes[2];          // n_nodes (batch array length)
    const int E = in_sizes[1] / 2;      // n_edges
    const int G = out_size / NCLS;      // n_graphs
    const int* src = ei;
    const int* dst = ei + E;

    // workspace carve-up
    float* xw   = (float*)d_ws;                 // M*128
    float* agg  = xw   + (size_t)M * FEAT;      // M*128
    float* dinv = agg  + (size_t)M * FEAT;      // M
    float* psum = dinv + M;                     // G*128
    float* pcnt = psum + (size_t)G * FEAT;      // G

    const int TB = 256;
    const int gM    = (M + TB - 1) / TB;
    const int gE    = (E + TB - 1) / TB;
    const long long nf = (long long)M * FEAT;
    const int gMF   = (int)((nf + TB - 1) / TB);
    const int gEdge = (int)(((long long)E * 32 + TB - 1) / TB);
    const int Mtiles = (M + 15) / 16;
    const int gGemm  = Mtiles < 1536 ? Mtiles : 1536;   // persistent blocks

    // dinv = rsqrt(1 + in-degree)
    k_init_deg   <<<gM, TB, 0, stream>>>(dinv, M);
    k_deg_count  <<<gE, TB, 0, stream>>>(dinv, dst, E);
    k_deg_to_dinv<<<gM, TB, 0, stream>>>(dinv, M);

    // layer 1: xw = x@W1 ; agg = D^-1/2 (A+I) D^-1/2 xw ; h1 = relu(agg+b1)
    k_gemm_f32_wmma<<<gGemm, TB, 0, stream>>>(x, W1, xw, M, Mtiles);
    k_self_init    <<<gMF,  TB, 0, stream>>>(agg, xw, dinv, M);
    k_edge_agg     <<<gEdge,TB, 0, stream>>>(agg, xw, dinv, src, dst, E);
    k_bias_act     <<<gMF,  TB, 0, stream>>>(agg, b1, M, 1);

    // layer 2: xw = h1@W2 ; agg = normalized aggregation ; h2 = agg+b2
    k_gemm_f32_wmma<<<gGemm, TB, 0, stream>>>(agg, W2, xw, M, Mtiles);
    k_self_init    <<<gMF,  TB, 0, stream>>>(agg, xw, dinv, M);  // reads xw only
    k_edge_agg     <<<gEdge,TB, 0, stream>>>(agg, xw, dinv, src, dst, E);
    k_bias_act     <<<gMF,  TB, 0, stream>>>(agg, b2, M, 0);

    // mean pool + classifier
    k_pool_zero <<<(G * FEAT + TB - 1) / TB, TB, 0, stream>>>(psum, pcnt, G);
    k_pool_accum<<<gMF, TB, 0, stream>>>(psum, pcnt, agg, batch, M);
    k_classify  <<<(G * NCLS + TB - 1) / TB, TB, 0, stream>>>(psum, pcnt, Wc, bc, out, G);
}